// skipgram_36197984370874
// MI455X (gfx1250) — compile-verified
//
#include <hip/hip_runtime.h>
#include <hip/hip_bf16.h>
#include <math.h>

// ---------------------------------------------------------------------------
// Skip-gram negative-sampling loss for MI455X (gfx1250).
//
// Per batch element b:
//   s_pos = dot(eu, ev) / 128
//   s_neg = dot(eu, sum_n neg_v[n]) / N        (mean of dots == dot of mean*N)
//   loss  = logsig(s_pos) + logsig(-s_neg)
// out = -sum_b loss / B
//
// Memory bound (~393 MB of row gathers); dots are done with exact-precision
// V_WMMA_F32_16X16X4_F32: 16 batch elements per wave, A rows = eu vectors,
// B cols = ev (or negsum) vectors, diagonal of the 16x16 D tile = the dots.
// ---------------------------------------------------------------------------

typedef __attribute__((ext_vector_type(2))) float v2f;
typedef __attribute__((ext_vector_type(8))) float v8f;

#define EMB_D            128
#define SLOTS_PER_WAVE   16
#define WAVES_PER_BLOCK  4
#define SLOTS_PER_BLOCK  (SLOTS_PER_WAVE * WAVES_PER_BLOCK)   // 64
#define LROW             132   // padded row stride in floats (16B aligned, bank-spread)

__device__ __forceinline__ float log_sigmoid(float x) {
    // log(sigmoid(x)) = min(x,0) - log1p(exp(-|x|))
    return fminf(x, 0.0f) - log1pf(expf(-fabsf(x)));
}

__global__ void sg_main(const float* __restrict__ uW,
                        const float* __restrict__ vW,
                        const int*   __restrict__ u_pos,
                        const int*   __restrict__ v_pos,
                        const int*   __restrict__ v_neg,
                        int B, int N,
                        float* __restrict__ blockSums) {
    extern __shared__ float smem[];
    float* s_eu = smem;                              // [64][LROW]
    float* s_ev = smem + SLOTS_PER_BLOCK * LROW;     // [64][LROW]
    float* s_w  = smem + 2 * SLOTS_PER_BLOCK * LROW; // [64][LROW] (sum of negatives)

    const int tid  = threadIdx.x;
    const int lane = tid & 31;
    const int wave = tid >> 5;
    const int blockSlot0 = blockIdx.x * SLOTS_PER_BLOCK;

    // ---- Phase 1: coalesced row gathers into LDS (one 512B row = 32 x float4)
    for (int i = 0; i < SLOTS_PER_WAVE; ++i) {
        const int sl = wave * SLOTS_PER_WAVE + i;       // local slot 0..63
        const int g  = blockSlot0 + sl;                 // global batch element
        const int gc = (g < B) ? g : 0;                 // clamp (loss masked later)

        const float4* up = (const float4*)(uW + (size_t)u_pos[gc] * EMB_D);
        float4 eu4 = up[lane];
        *(float4*)&s_eu[sl * LROW + lane * 4] = eu4;

        const float4* vp = (const float4*)(vW + (size_t)v_pos[gc] * EMB_D);
        float4 ev4 = vp[lane];
        *(float4*)&s_ev[sl * LROW + lane * 4] = ev4;

        float4 acc = make_float4(0.f, 0.f, 0.f, 0.f);
        const int* np = v_neg + (size_t)gc * N;
        for (int n = 0; n < N; ++n) {
            const float4* rp = (const float4*)(vW + (size_t)np[n] * EMB_D);
            float4 r = rp[lane];
            acc.x += r.x; acc.y += r.y; acc.z += r.z; acc.w += r.w;
        }
        *(float4*)&s_w[sl * LROW + lane * 4] = acc;
    }
    __syncthreads();

    // ---- Phase 2: fp32 WMMA 16x16x4, accumulate over K = 128 in chunks of 4.
    // A (16x4): lane = h*16+s holds {A[s][2h], A[s][2h+1]}
    // B (4x16): lane = h*16+n holds {B[2h][n], B[2h+1][n]}
    // -> both operands fetch float2 at row (lane&15), col 4*kc + 2*(lane>>4).
    const int s = lane & 15;
    const int h = lane >> 4;
    const int rowBase = (wave * SLOTS_PER_WAVE + s) * LROW + 2 * h;

    v8f accP = {};   // diag -> dot(eu, ev)
    v8f accN = {};   // diag -> dot(eu, negsum)
#pragma unroll
    for (int kc = 0; kc < EMB_D / 4; ++kc) {
        const int off = rowBase + 4 * kc;
        v2f a  = *(const v2f*)&s_eu[off];
        v2f bP = *(const v2f*)&s_ev[off];
        v2f bN = *(const v2f*)&s_w[off];
        accP = __builtin_amdgcn_wmma_f32_16x16x4_f32(
            false, a, false, bP, (short)0, accP, false, false);
        accN = __builtin_amdgcn_wmma_f32_16x16x4_f32(
            false, a, false, bN, (short)0, accN, false, false);
    }

    // ---- Diagonal extraction.
    // D tile layout: VGPR r, lanes 0-15 -> (M=r, N=lane); lanes 16-31 -> (M=r+8, N=lane-16).
    // Diag (m,m): m=0..7 -> lane m, acc[m];  m=8..15 -> lane m+16, acc[m-8].
    const bool active = (lane < 8) || (lane >= 24);
    const int  r      = (lane < 8) ? lane : (lane - 24);
    float dP = accP[0], dN = accN[0];
#pragma unroll
    for (int j = 1; j < 8; ++j) {
        if (r == j) { dP = accP[j]; dN = accN[j]; }
    }

    const int slot = wave * SLOTS_PER_WAVE + ((lane < 8) ? lane : (lane - 16));
    const int g    = blockSlot0 + slot;
    float loss = 0.0f;
    if (active && g < B) {
        const float sp = dP * (1.0f / (float)EMB_D);
        const float sn = dN * (1.0f / (float)N);
        loss = log_sigmoid(sp) + log_sigmoid(-sn);
    }

    // ---- Deterministic per-block reduction (each of 64 slots written once).
    __syncthreads();
    float* red = smem;
    if (active) red[slot] = loss;
    __syncthreads();
    if (tid == 0) {
        float sum = 0.0f;
        for (int i = 0; i < SLOTS_PER_BLOCK; ++i) sum += red[i];
        blockSums[blockIdx.x] = sum;
    }
}

__global__ void sg_reduce(const float* __restrict__ blockSums, int nBlocks, int B,
                          float* __restrict__ out) {
    __shared__ float sdata[256];
    float sum = 0.0f;
    for (int i = threadIdx.x; i < nBlocks; i += 256) sum += blockSums[i];
    sdata[threadIdx.x] = sum;
    __syncthreads();
    for (int stride = 128; stride > 0; stride >>= 1) {
        if ((int)threadIdx.x < stride) sdata[threadIdx.x] += sdata[threadIdx.x + stride];
        __syncthreads();
    }
    if (threadIdx.x == 0) out[0] = -sdata[0] / (float)B;
}

extern "C" void kernel_launch(void* const* d_in, const int* in_sizes, int n_in,
                              void* d_out, int out_size, void* d_ws, size_t ws_size,
                              hipStream_t stream) {
    const float* uW    = (const float*)d_in[0];
    const float* vW    = (const float*)d_in[1];
    const int*   u_pos = (const int*)d_in[2];
    const int*   v_pos = (const int*)d_in[3];
    const int*   v_neg = (const int*)d_in[4];

    const int B = in_sizes[2];
    const int N = in_sizes[4] / B;
    const int nBlocks = (B + SLOTS_PER_BLOCK - 1) / SLOTS_PER_BLOCK;

    float* blockSums = (float*)d_ws;
    const size_t shmem = (size_t)3 * SLOTS_PER_BLOCK * LROW * sizeof(float);

    sg_main<<<nBlocks, WAVES_PER_BLOCK * 32, shmem, stream>>>(
        uW, vW, u_pos, v_pos, v_neg, B, N, blockSums);
    sg_reduce<<<1, 256, 0, stream>>>(blockSums, nBlocks, B, (float*)d_out);
}